// SelfAttention_43095701848751
// MI455X (gfx1250) — compile-verified
//
#include <hip/hip_runtime.h>
#include <stdint.h>

#define BB 64
#define SS 2048
#define DD 512
#define RS (DD + 4)      // padded LDS row stride (floats): 516 % 64 banks = 4 -> conflict-free
#define NC 16            // S-chunks per batch row
#define CH (SS / NC)     // 128 positions per chunk
#define GP 16            // positions per group = WMMA tile M
#define NG (CH / GP)     // 8 groups per chunk
#define THREADS 256
#define WSSTRIDE (DD + 2)

typedef __attribute__((ext_vector_type(2))) float v2f;
typedef __attribute__((ext_vector_type(8))) float v8f;

__global__ __launch_bounds__(THREADS) void attn_partial_kernel(
    const float* __restrict__ inp, const int* __restrict__ lens,
    const float* __restrict__ w, const float* __restrict__ bias,
    float* __restrict__ ws)
{
    __shared__ float lds_tile[2][GP * RS];   // 2 x ~32.25KB double-buffered inp tiles
    __shared__ float lds_w[DD];              // 2KB scorer weights
    __shared__ float lds_scores[GP];
    __shared__ float lds_p[GP];

    const int c    = blockIdx.x;     // S-chunk index
    const int b    = blockIdx.y;     // batch row
    const int tid  = threadIdx.x;
    const int lane = tid & 31;
    const int wv   = tid >> 5;       // wave 0..7

    // Load scorer weights into LDS (visible after first barrier in the loop).
    lds_w[tid]       = w[tid];
    lds_w[tid + 256] = w[tid + 256];
    const float bias_v = bias[0];
    const int   len    = lens[b];
    const int   s0     = c * CH;

    // 64-bit batch base for async loads (per-batch span is 4MB -> 32-bit voffset ok)
    const uint64_t base_u = (uint64_t)(uintptr_t)(inp + (size_t)b * SS * DD);
    const uint32_t tile_a = (uint32_t)(uintptr_t)(&lds_tile[0][0]);
    const uint32_t tile_b = (uint32_t)(uintptr_t)(&lds_tile[1][0]);

    // Async-copy one 16x512 f32 tile (32KB) into padded LDS rows:
    // thread owns 8 (row, 16B-column) transfers; global stride 2048B, LDS stride 2064B.
    const uint32_t colb  = (uint32_t)(tid & 127) * 16u;  // byte column within a row
    const int      rbase = (tid >> 7) * 8;               // rows [rbase, rbase+8)
    auto issue_group = [&](int g, uint32_t lds_base) {
        const uint32_t gbase = (uint32_t)((s0 + g * GP) * DD * 4);
        #pragma unroll
        for (int i = 0; i < 8; ++i) {
            const int row = rbase + i;
            uint32_t goff  = gbase + (uint32_t)row * (DD * 4) + colb;
            uint32_t laddr = lds_base + (uint32_t)row * (RS * 4) + colb;
            asm volatile("global_load_async_to_lds_b128 %0, %1, %2 offset:0"
                         :: "v"(laddr), "v"(goff), "s"(base_u) : "memory");
        }
    };

    float m = -3.0e38f, denom = 0.f, acc0 = 0.f, acc1 = 0.f;
    const int d0 = tid, d1 = tid + 256;

    issue_group(0, tile_a);

    for (int g = 0; g < NG; ++g) {
        const float* curp = (g & 1) ? &lds_tile[1][0] : &lds_tile[0][0];
        if (g + 1 < NG) {
            issue_group(g + 1, (g & 1) ? tile_a : tile_b);
            asm volatile("s_wait_asynccnt 8" ::: "memory");  // current tile done
        } else {
            asm volatile("s_wait_asynccnt 0" ::: "memory");
        }
        __syncthreads();                       // tile visible; prev consumption done
        if (tid < GP) lds_scores[tid] = bias_v;
        __syncthreads();

        // ---- WMMA scores: wave wv covers K in [wv*64, wv*64+64) ----
        {
            const int mrow  = lane & 15;
            const int koff2 = (lane < 16) ? 0 : 2;   // f32 A layout: K = {0,1} / {2,3}
            const float* arow = curp + mrow * RS;
            v8f cacc = {0.f, 0.f, 0.f, 0.f, 0.f, 0.f, 0.f, 0.f};
            const int kbase = wv * 64 + koff2;
            #pragma unroll
            for (int kk = 0; kk < 64; kk += 4) {
                const int k = kbase + kk;
                v2f av; av.x = arow[k];  av.y = arow[k + 1];
                v2f bv; bv.x = lds_w[k]; bv.y = lds_w[k + 1]; // w replicated over N cols
                cacc = __builtin_amdgcn_wmma_f32_16x16x4_f32(
                    false, av, false, bv, (short)0, cacc, false, false);
            }
            // All N columns equal; lanes 0 and 16 hold M=0..7 / M=8..15 in c[0..7].
            if ((lane & 15) == 0) {
                const int mb = (lane >> 4) * 8;
                #pragma unroll
                for (int r = 0; r < 8; ++r)
                    atomicAdd(&lds_scores[mb + r], cacc[r]);
            }
        }
        __syncthreads();                       // scores final

        // ---- group-local softmax factors (identical across threads) ----
        const int sg0 = s0 + g * GP;
        float gm = -3.0e38f;
        #pragma unroll
        for (int j = 0; j < GP; ++j)
            if (sg0 + j < len) gm = fmaxf(gm, lds_scores[j]);
        if (tid < GP) {
            float pj = 0.f;
            if ((sg0 + tid < len) && (gm > -1.0e38f))
                pj = __expf(lds_scores[tid] - gm);
            lds_p[tid] = pj;
        }
        __syncthreads();                       // p ready

        float gsum = 0.f, a0 = 0.f, a1 = 0.f;
        #pragma unroll
        for (int j = 0; j < GP; ++j) {
            const float pj = lds_p[j];
            gsum += pj;
            a0 += pj * curp[j * RS + d0];
            a1 += pj * curp[j * RS + d1];
        }
        const float nm = fmaxf(m, gm);
        const float fo = __expf(m - nm);       // old-scale (0 if m was -inf sentinel)
        const float fn = __expf(gm - nm);
        denom = denom * fo + gsum * fn;
        acc0  = acc0  * fo + a0   * fn;
        acc1  = acc1  * fo + a1   * fn;
        m = nm;
        __syncthreads();                       // done before buffer is overwritten
    }

    float* wsp = ws + (size_t)(b * NC + c) * WSSTRIDE;
    if (tid == 0) { wsp[0] = m; wsp[1] = denom; }
    wsp[2 + d0] = acc0;
    wsp[2 + d1] = acc1;
}

__global__ __launch_bounds__(THREADS) void attn_combine_kernel(
    const float* __restrict__ ws, float* __restrict__ out)
{
    const int b = blockIdx.x;
    const int t = threadIdx.x;
    const float* base = ws + (size_t)b * NC * WSSTRIDE;

    float M = -3.0e38f;
    #pragma unroll
    for (int c = 0; c < NC; ++c) M = fmaxf(M, base[c * WSSTRIDE]);

    float total = 0.f, n0 = 0.f, n1 = 0.f;
    #pragma unroll
    for (int c = 0; c < NC; ++c) {
        const float* p = base + c * WSSTRIDE;
        const float wgt = __expf(p[0] - M);    // 0 for fully-masked chunks
        total += wgt * p[1];
        n0 += wgt * p[2 + t];
        n1 += wgt * p[2 + 256 + t];
    }
    out[b * DD + t]       = n0 / total;        // total > 0 since len >= 1
    out[b * DD + 256 + t] = n1 / total;
}

extern "C" void kernel_launch(void* const* d_in, const int* in_sizes, int n_in,
                              void* d_out, int out_size, void* d_ws, size_t ws_size,
                              hipStream_t stream) {
    const float* inp  = (const float*)d_in[0];
    const int*   lens = (const int*)d_in[1];
    const float* w    = (const float*)d_in[2];
    const float* bias = (const float*)d_in[3];
    float* ws  = (float*)d_ws;    // needs 64*16*514*4 = ~2.1 MB
    float* out = (float*)d_out;

    attn_partial_kernel<<<dim3(NC, BB), THREADS, 0, stream>>>(inp, lens, w, bias, ws);
    attn_combine_kernel<<<dim3(BB), THREADS, 0, stream>>>(ws, out);
}